// HierarchicalCategoricalLogLikelihoodLossLayer_65189013618799
// MI455X (gfx1250) — compile-verified
//
#include <hip/hip_runtime.h>
#include <hip/hip_bf16.h>
#include <math.h>

// ---------------- problem constants ----------------
#define BB 32
#define TT 1024
#define KK 16
#define DD 80
#define PP 64
#define TTILE 64              // T columns per block in the WMMA GEMM
#define HALF_LOG2PI_D 73.5150850522431f   // 0.5 * 80 * ln(2*pi)

typedef __attribute__((ext_vector_type(2))) float v2f;
typedef __attribute__((ext_vector_type(8))) float v8f;

// workspace layout (in floats)
static constexpr size_t OFS_LINV   = 0;                         // 512*6400
static constexpr size_t OFS_LOGDET = OFS_LINV + (size_t)BB*KK*DD*DD; // 512
static constexpr size_t OFS_MAHA   = OFS_LOGDET + (size_t)BB*KK;     // 512*1024
static constexpr size_t OFS_LLP    = OFS_MAHA + (size_t)BB*KK*TT;    // 128
static constexpr size_t OFS_REGP   = OFS_LLP + 128;                  // 1024

// ---------------------------------------------------------------------------
// Kernel 1: per (b,k) invert tril(L_subj[sid[b],k]) by forward substitution,
// and compute logdet = sum(log(diag)).  One 128-thread block per (b,k);
// each of 80 threads solves one column independently (no cross-thread deps).
// ---------------------------------------------------------------------------
__global__ __launch_bounds__(128) void k1_invert(const float* __restrict__ L_subj,
                                                 const int* __restrict__ sid,
                                                 float* __restrict__ Linv,
                                                 float* __restrict__ logdet)
{
    __shared__ float sL[DD * 81];
    __shared__ float sX[DD * 81];
    const int bk = blockIdx.x;            // 0..511
    const int k  = bk & (KK - 1);
    const int b  = bk >> 4;
    const int tid = threadIdx.x;
    const int p = sid[b];

    const float* src = L_subj + ((size_t)(p * KK + k)) * (DD * DD);
    for (int i = tid; i < DD * DD; i += 128) {
        int r = i / DD, c = i - r * DD;
        sL[r * 81 + c] = (c <= r) ? src[i] : 0.0f;   // tril
    }
    __syncthreads();

    if (tid < DD) {
        const int j = tid;                // column of the inverse
        for (int i = j; i < DD; ++i) {
            float sum = (i == j) ? 1.0f : 0.0f;
            for (int m = j; m < i; ++m) sum -= sL[i * 81 + m] * sX[m * 81 + j];
            sX[i * 81 + j] = sum / sL[i * 81 + i];
        }
    }
    __syncthreads();

    float* dst = Linv + (size_t)bk * (DD * DD);
    for (int i = tid; i < DD * DD; i += 128) {
        int r = i / DD, c = i - r * DD;
        dst[i] = (c <= r) ? sX[r * 81 + c] : 0.0f;
    }
    if (tid == 0) {
        float ld = 0.0f;
        for (int d = 0; d < DD; ++d) ld += __logf(sL[d * 81 + d]);
        logdet[bk] = ld;
    }
}

// ---------------------------------------------------------------------------
// Kernel 2: WMMA fp32 GEMM  Y = Linv(80x80) * Diff(80x64-tile), folding the
// row reduction maha[t] = sum_d Y[d][t]^2 into registers + one lane-swap.
// Block = 128 threads = 4 waves; wave w owns N-tile columns [w*16, w*16+16).
// A/B fragments for V_WMMA_F32_16X16X4_F32 are float2 per lane:
//   A: lane (l15,hi) holds A[M=l15][K = kk + 2*hi + {0,1}]
//   B: lane (l15,hi) holds B[K = kk + 2*hi + {0,1}][N = l15]
// Rows padded to 82 floats so each fragment is one aligned ds_load_b64.
// ---------------------------------------------------------------------------
__global__ __launch_bounds__(128) void k2_wmma_maha(const float* __restrict__ x,
                                                    const float* __restrict__ Linv,
                                                    const float* __restrict__ mu_subj,
                                                    const int* __restrict__ sid,
                                                    float* __restrict__ maha)
{
    __shared__ float sL[DD * 82];        // Linv, row-major, pad 82
    __shared__ float sD[TTILE * 82];     // diff transposed: sD[t][d], pad 82
    __shared__ float sMu[DD];

    const int bid = blockIdx.x;          // 32*16*16 = 8192 blocks
    const int tt  = bid & 15;            // T-tile index (T/TTILE = 16)
    const int k   = (bid >> 4) & (KK - 1);
    const int b   = bid >> 8;
    const int tid = threadIdx.x;
    const int p = sid[b];

    if (tid < DD) sMu[tid] = mu_subj[((size_t)(p * KK + k)) * DD + tid];
    __syncthreads();

    const float* Lsrc = Linv + (size_t)(b * KK + k) * (DD * DD);
    for (int i = tid; i < DD * DD; i += 128) {
        int r = i / DD, c = i - r * DD;
        sL[r * 82 + c] = Lsrc[i];
    }
    const float* xsrc = x + ((size_t)b * TT + (size_t)tt * TTILE) * DD;
    for (int e = tid; e < TTILE * DD; e += 128) {
        int t = e / DD, d = e - t * DD;
        sD[t * 82 + d] = xsrc[e] - sMu[d];
    }
    __syncthreads();

    const int lane = tid & 31;
    const int wave = tid >> 5;           // 0..3
    const int l15  = lane & 15;
    const int hi   = lane >> 4;          // 0 or 1
    const int n0   = wave * 16;

    v8f acc[5] = {};                     // 5 M-tiles cover D rows 0..79
    #pragma unroll 4
    for (int kk = 0; kk < DD; kk += 4) {
        const int colA = kk + 2 * hi;
        v2f bfrag = *(const v2f*)&sD[(n0 + l15) * 82 + colA];
        #pragma unroll
        for (int m = 0; m < 5; ++m) {
            v2f afrag = *(const v2f*)&sL[(m * 16 + l15) * 82 + colA];
            acc[m] = __builtin_amdgcn_wmma_f32_16x16x4_f32(
                false, afrag, false, bfrag, (short)0, acc[m], false, false);
        }
    }

    // maha[t] = sum over all 80 rows of Y^2.  Lane holds rows {m*16+r+8*hi}.
    float s = 0.0f;
    #pragma unroll
    for (int m = 0; m < 5; ++m) {
        #pragma unroll
        for (int r = 0; r < 8; ++r) { float v = acc[m][r]; s += v * v; }
    }
    s += __shfl_xor(s, 16, 32);          // combine M and M+8 halves (wave32)
    if (lane < 16) {
        int t = tt * TTILE + n0 + l15;
        maha[((size_t)(b * KK + k)) * TT + t] = s;
    }
}

// ---------------------------------------------------------------------------
// Kernel 3a: weighted NLL partials.  128 blocks x 256 threads = 32768 (b,t).
// acc = sum_k gamma[b,t,k] * (C + logdet[b,k] + 0.5*maha[b,k,t])  == -ll_bt
// Deterministic shared-memory tree reduction per block.
// ---------------------------------------------------------------------------
__global__ __launch_bounds__(256) void k3a_ll(const float* __restrict__ gamma,
                                              const float* __restrict__ logdet,
                                              const float* __restrict__ maha,
                                              float* __restrict__ llp)
{
    __shared__ float red[256];
    const int tid = threadIdx.x;
    const int bt  = blockIdx.x * 256 + tid;
    const int b   = bt >> 10;            // / 1024
    const int t   = bt & 1023;

    float acc = 0.0f;
    #pragma unroll
    for (int k = 0; k < KK; ++k) {
        float g  = gamma[((size_t)b * TT + t) * KK + k];
        float ld = logdet[b * KK + k];
        float mh = maha[((size_t)(b * KK + k)) * TT + t];
        acc += g * (HALF_LOG2PI_D + ld + 0.5f * mh);
    }
    red[tid] = acc;
    __syncthreads();
    for (int s = 128; s > 0; s >>= 1) {
        if (tid < s) red[tid] += red[tid + s];
        __syncthreads();
    }
    if (tid == 0) llp[blockIdx.x] = red[0];
}

// ---------------------------------------------------------------------------
// Kernel 3r: regularization partials per (p,k).  1024 blocks x 256 threads.
// partial = mask[p] * 0.05 * ( sum(mu diff^2) + sum(L diff^2) )  for slice k.
// ---------------------------------------------------------------------------
__global__ __launch_bounds__(256) void k3r_reg(const float* __restrict__ mu_subj,
                                               const float* __restrict__ mu_pop,
                                               const float* __restrict__ L_subj,
                                               const float* __restrict__ L_pop,
                                               const int* __restrict__ sid,
                                               float* __restrict__ regp)
{
    __shared__ float red[256];
    const int pk = blockIdx.x;           // p*16 + k
    const int k  = pk & (KK - 1);
    const int p  = pk >> 4;
    const int tid = threadIdx.x;

    bool present = false;
    for (int i = 0; i < BB; ++i) present |= (sid[i] == p);

    float acc = 0.0f;
    const float* Ls = L_subj + (size_t)pk * (DD * DD);
    const float* Lp = L_pop + (size_t)k * (DD * DD);
    for (int i = tid; i < DD * DD; i += 256) {
        float d = Ls[i] - Lp[i];
        acc += d * d;
    }
    const float* ms = mu_subj + (size_t)pk * DD;
    const float* mp = mu_pop + (size_t)k * DD;
    for (int i = tid; i < DD; i += 256) {
        float d = ms[i] - mp[i];
        acc += d * d;
    }
    red[tid] = acc;
    __syncthreads();
    for (int s = 128; s > 0; s >>= 1) {
        if (tid < s) red[tid] += red[tid + s];
        __syncthreads();
    }
    if (tid == 0) regp[pk] = present ? 0.05f * red[0] : 0.0f;  // lambda/2 = 0.05
}

// ---------------------------------------------------------------------------
// Kernel 3f: single-thread deterministic finisher.
// ---------------------------------------------------------------------------
__global__ void k3f_final(const float* __restrict__ llp,
                          const float* __restrict__ regp,
                          const int* __restrict__ sid,
                          float* __restrict__ out)
{
    float nll = 0.0f;
    for (int i = 0; i < 128; ++i) nll += llp[i];
    nll *= (1.0f / (float)(BB * TT));
    float rg = 0.0f;
    for (int i = 0; i < PP * KK; ++i) rg += regp[i];
    int nu = 0;
    for (int p = 0; p < PP; ++p) {
        bool f = false;
        for (int i = 0; i < BB; ++i) f |= (sid[i] == p);
        nu += f ? 1 : 0;
    }
    out[0] = nll + ((float)nu / (float)PP) * rg;
}

// ---------------------------------------------------------------------------
extern "C" void kernel_launch(void* const* d_in, const int* in_sizes, int n_in,
                              void* d_out, int out_size, void* d_ws, size_t ws_size,
                              hipStream_t stream) {
    (void)in_sizes; (void)n_in; (void)out_size; (void)ws_size;
    const float* x       = (const float*)d_in[0];
    const float* mu_pop  = (const float*)d_in[1];
    const float* L_pop   = (const float*)d_in[2];
    const float* mu_subj = (const float*)d_in[3];
    const float* L_subj  = (const float*)d_in[4];
    const float* gamma   = (const float*)d_in[5];
    const int*   sid     = (const int*)d_in[6];

    float* ws     = (float*)d_ws;
    float* Linv   = ws + OFS_LINV;
    float* logdet = ws + OFS_LOGDET;
    float* maha   = ws + OFS_MAHA;
    float* llp    = ws + OFS_LLP;
    float* regp   = ws + OFS_REGP;

    k1_invert<<<BB * KK, 128, 0, stream>>>(L_subj, sid, Linv, logdet);
    k2_wmma_maha<<<BB * KK * (TT / TTILE), 128, 0, stream>>>(x, Linv, mu_subj, sid, maha);
    k3a_ll<<<(BB * TT) / 256, 256, 0, stream>>>(gamma, logdet, maha, llp);
    k3r_reg<<<PP * KK, 256, 0, stream>>>(mu_subj, mu_pop, L_subj, L_pop, sid, regp);
    k3f_final<<<1, 1, 0, stream>>>(llp, regp, sid, (float*)d_out);
}